// CausalSelfAttention_26826365731091
// MI455X (gfx1250) — compile-verified
//
#include <hip/hip_runtime.h>
#include <hip/hip_bf16.h>

typedef float v2f __attribute__((ext_vector_type(2)));
typedef float v8f __attribute__((ext_vector_type(8)));

#define B_ 4
#define T_ 2048
#define D_ 1024
#define H_ 16
#define HD_ 64
#define M_ (B_ * T_)                      // 8192 rows
#define QKV_STRIDE ((size_t)B_ * H_ * T_ * HD_)   // 8388608 floats per Q/K/V
#define QTILES (T_ / 16)                  // 128
#define TILES (B_ * H_ * QTILES)          // 8192 flash work tiles

static __device__ __forceinline__ v8f wmma_f32(v2f a, v2f b, v8f c) {
  // D = A(16x4 f32) * B(4x16 f32) + C(16x16 f32), full fp32 precision
  return __builtin_amdgcn_wmma_f32_16x16x4_f32(false, a, false, b, (short)0, c,
                                               false, false);
}

// ---------------------------------------------------------------------------
// GEMM: out[m][n] = sum_k A[m][k] * W[n][k]   (i.e. A @ W^T)
// Block: 256 threads = 8 waves (2 x 4), block tile 64(M) x 128(N), wave 32x32.
// mode 0: dense store to out[m*N+n]
// mode 1: scatter into Q/K/V workspace laid out [3][B][H][T][HD]
// ---------------------------------------------------------------------------
__global__ __launch_bounds__(256) void gemm_xwt(const float* __restrict__ A,
                                                const float* __restrict__ W,
                                                float* __restrict__ out,
                                                float* __restrict__ qkv,
                                                int M, int N, int K, int mode) {
  __shared__ float As[64][20];    // [m][k], stride 20 -> conflict-free, 16B ok
  __shared__ float Bs[128][20];   // [n][k]

  const int tid  = threadIdx.x;
  const int lane = tid & 31;
  const int wave = tid >> 5;
  const int wr   = wave >> 2;          // 0..1  (M direction)
  const int wc   = wave & 3;           // 0..3  (N direction)
  const int l16  = lane & 15;
  const int half = lane >> 4;          // 0/1
  const int kh   = half << 1;          // k offset within 4-wide step
  const int m0   = blockIdx.y * 64;
  const int n0   = blockIdx.x * 128;

  const v8f vz = {0.f, 0.f, 0.f, 0.f, 0.f, 0.f, 0.f, 0.f};
  v8f acc[2][2];
#pragma unroll
  for (int i = 0; i < 2; ++i)
#pragma unroll
    for (int j = 0; j < 2; ++j) acc[i][j] = vz;

  for (int k0 = 0; k0 < K; k0 += 16) {
    // --- cooperative LDS fill ---
    {
      const int row = tid >> 2, q4 = (tid & 3) << 2;
      const float4 av = *(const float4*)(A + (size_t)(m0 + row) * K + k0 + q4);
      *(float4*)&As[row][q4] = av;
#pragma unroll
      for (int i = 0; i < 2; ++i) {
        const int idx  = tid + i * 256;
        const int brow = idx >> 2, bq4 = (idx & 3) << 2;
        const float4 bv =
            *(const float4*)(W + (size_t)(n0 + brow) * K + k0 + bq4);
        *(float4*)&Bs[brow][bq4] = bv;
      }
    }
    __syncthreads();

    // --- 16 WMMAs per wave per k-chunk ---
#pragma unroll
    for (int kb = 0; kb < 4; ++kb) {
      v2f af[2], bf[2];
#pragma unroll
      for (int i = 0; i < 2; ++i)
        af[i] = *(const v2f*)&As[wr * 32 + i * 16 + l16][kb * 4 + kh];
#pragma unroll
      for (int j = 0; j < 2; ++j)
        bf[j] = *(const v2f*)&Bs[wc * 32 + j * 16 + l16][kb * 4 + kh];
#pragma unroll
      for (int i = 0; i < 2; ++i)
#pragma unroll
        for (int j = 0; j < 2; ++j)
          acc[i][j] = wmma_f32(af[i], bf[j], acc[i][j]);
    }
    __syncthreads();
  }

  // --- epilogue: C layout lane(L) vgpr(r) -> (M = r + 8*half, N = l16) ---
#pragma unroll
  for (int i = 0; i < 2; ++i)
#pragma unroll
    for (int j = 0; j < 2; ++j)
#pragma unroll
      for (int r = 0; r < 8; ++r) {
        const int m = m0 + wr * 32 + i * 16 + r + (half << 3);
        const int n = n0 + wc * 32 + j * 16 + l16;
        const float v = acc[i][j][r];
        if (mode == 0) {
          out[(size_t)m * N + n] = v;
        } else {
          const int which = n >> 10;        // 0=Q 1=K 2=V (H*HD == 1024)
          const int rem   = n & 1023;
          const int hh    = rem >> 6;
          const int dd    = rem & 63;
          const int bb    = m >> 11;        // T == 2048
          const int tt    = m & 2047;
          qkv[(size_t)which * QKV_STRIDE +
              (((size_t)(bb * H_ + hh)) * T_ + tt) * HD_ + dd] = v;
        }
      }
}

// ---------------------------------------------------------------------------
// Flash attention, causal. One wave owns a 16-row query tile; online softmax.
// Q/K/V laid out [B][H][T][HD]. Output written as [B][T][H*HD] for proj GEMM.
// ---------------------------------------------------------------------------
__global__ __launch_bounds__(128) void flash_causal(
    const float* __restrict__ Q, const float* __restrict__ Kv,
    const float* __restrict__ V, const unsigned char* __restrict__ amask,
    float* __restrict__ O) {
  __shared__ float Ps[4][16][18];   // per-wave P-tile staging (transpose)

  const int wave = threadIdx.x >> 5;
  const int lane = threadIdx.x & 31;
  const int l16  = lane & 15;
  const int half = lane >> 4;
  const int tile = blockIdx.x * 4 + wave;     // 0..TILES-1
  const int qt   = tile & (QTILES - 1);       // query tile in [0,128)
  const int bh   = tile >> 7;                 // b*H + h
  const int b    = bh >> 4;
  const int hh   = bh & 15;

  const float* qp = Q + ((size_t)bh * T_ + (size_t)qt * 16) * HD_;
  const float* kp = Kv + (size_t)bh * T_ * HD_;
  const float* vp = V + (size_t)bh * T_ * HD_;

  // Q fragments resident in registers: A layout, lane -> (M=l16, K=kb*4+2*half..+1)
  v2f qf[16];
#pragma unroll
  for (int kb = 0; kb < 16; ++kb)
    qf[kb] = *(const v2f*)(qp + (size_t)l16 * HD_ + kb * 4 + half * 2);

  const v8f vz = {0.f, 0.f, 0.f, 0.f, 0.f, 0.f, 0.f, 0.f};
  v8f o[4];
#pragma unroll
  for (int j = 0; j < 4; ++j) o[j] = vz;
  float mrow[8], lrow[8], alpha[8];
#pragma unroll
  for (int r = 0; r < 8; ++r) {
    mrow[r] = -1e30f;
    lrow[r] = 0.f;
  }

  const int q0 = qt * 16;
  float(*ps)[18] = Ps[wave];

  for (int kt = 0; kt <= qt; ++kt) {
    // S = (Q @ K^T) for this 16x16 tile: 16 fp32 WMMAs over hd=64
    v8f s = vz;
#pragma unroll
    for (int kb = 0; kb < 16; ++kb) {
      const v2f bf =
          *(const v2f*)(kp + (size_t)(kt * 16 + l16) * HD_ + kb * 4 + half * 2);
      s = wmma_f32(qf[kb], bf, s);
    }

    const int key = kt * 16 + l16;
    const bool kvalid = amask[b * T_ + key] != 0;

    // scale + causal/attention mask + online softmax (rows live in 16-lane halves)
#pragma unroll
    for (int r = 0; r < 8; ++r) {
      const int qrow = q0 + r + half * 8;
      float sv = s[r] * 0.125f;                       // 1/sqrt(64)
      sv = ((key <= qrow) && kvalid) ? sv : -1e30f;
      float mx = sv;
#pragma unroll
      for (int off = 1; off < 16; off <<= 1)
        mx = fmaxf(mx, __shfl_xor(mx, off, 16));
      const float mnew = fmaxf(mrow[r], mx);
      alpha[r] = __expf(mrow[r] - mnew);
      mrow[r] = mnew;
      const float p = __expf(sv - mnew);
      s[r] = p;
      float rs = p;
#pragma unroll
      for (int off = 1; off < 16; off <<= 1) rs += __shfl_xor(rs, off, 16);
      lrow[r] = lrow[r] * alpha[r] + rs;
    }
#pragma unroll
    for (int j = 0; j < 4; ++j)
#pragma unroll
      for (int r = 0; r < 8; ++r) o[j][r] *= alpha[r];

    // Transpose P (C layout) -> A layout via per-wave LDS tile
#pragma unroll
    for (int r = 0; r < 8; ++r) ps[r + half * 8][l16] = s[r];
    asm volatile("s_wait_dscnt 0" ::: "memory");
    __builtin_amdgcn_wave_barrier();

    // O += P @ V : 16 fp32 WMMAs (4 k-steps x 4 hd-subtiles)
#pragma unroll
    for (int kb = 0; kb < 4; ++kb) {
      const v2f pa = *(const v2f*)&ps[l16][kb * 4 + half * 2];
#pragma unroll
      for (int j = 0; j < 4; ++j) {
        const float* vrow =
            vp + (size_t)(kt * 16 + kb * 4 + half * 2) * HD_ + j * 16 + l16;
        v2f vb;
        vb.x = vrow[0];
        vb.y = vrow[HD_];
        o[j] = wmma_f32(pa, vb, o[j]);
      }
    }
    asm volatile("s_wait_dscnt 0" ::: "memory");
    __builtin_amdgcn_wave_barrier();
  }

  // normalize and store as [B][T][H*HD]
#pragma unroll
  for (int j = 0; j < 4; ++j)
#pragma unroll
    for (int r = 0; r < 8; ++r) {
      const int qrow = q0 + r + half * 8;
      O[((size_t)b * T_ + qrow) * D_ + hh * HD_ + j * 16 + l16] =
          o[j][r] / lrow[r];
    }
}

// ---------------------------------------------------------------------------
extern "C" void kernel_launch(void* const* d_in, const int* in_sizes, int n_in,
                              void* d_out, int out_size, void* d_ws,
                              size_t ws_size, hipStream_t stream) {
  (void)in_sizes; (void)n_in; (void)out_size; (void)ws_size;
  const float* x               = (const float*)d_in[0];
  const unsigned char* amask   = (const unsigned char*)d_in[1];
  const float* w_qkv           = (const float*)d_in[2];
  const float* w_proj          = (const float*)d_in[3];
  float* out = (float*)d_out;
  float* ws  = (float*)d_ws;

  float* Qm = ws;                         // [B][H][T][HD]
  float* Km = ws + QKV_STRIDE;
  float* Vm = ws + 2 * QKV_STRIDE;
  float* AO = ws + 3 * QKV_STRIDE;        // [B][T][D]

  // 1) qkv = x @ w_qkv^T, scattered into Q/K/V head-major layout
  gemm_xwt<<<dim3((3 * D_) / 128, M_ / 64), 256, 0, stream>>>(
      x, w_qkv, nullptr, ws, M_, 3 * D_, D_, 1);

  // 2) causal flash attention
  flash_causal<<<dim3(TILES / 4), 128, 0, stream>>>(Qm, Km, Vm, amask, AO);

  // 3) out = attn_out @ w_proj^T
  gemm_xwt<<<dim3(D_ / 128, M_ / 64), 256, 0, stream>>>(
      AO, w_proj, out, nullptr, M_, D_, D_, 0);
}